// MultiScaleLoss_79783312490501
// MI455X (gfx1250) — compile-verified
//
#include <hip/hip_runtime.h>

typedef __attribute__((ext_vector_type(2))) float v2f;
typedef __attribute__((ext_vector_type(8))) float v8f;

#define A_DIM 512
#define NPIX (A_DIM * A_DIM)        // 262144 = 2^18
#define BATCH 8
#define NE (BATCH * NPIX)           // 2097152 = 2^21 elements per scale map
#define NUM_SCALES 7
#define NBLK 512
#define TPB 256
#define NTHREADS (NBLK * TPB)       // 131072 = 2^17
#define EPT (NE / NTHREADS)         // 16 elements per thread, exact

// ---------------- wave reductions (wave32) ----------------
__device__ __forceinline__ float wave_sum32(float v) {
#pragma unroll
  for (int m = 16; m >= 1; m >>= 1) v += __shfl_xor(v, m, 32);
  return v;
}
__device__ __forceinline__ float wave_max32(float v) {
#pragma unroll
  for (int m = 16; m >= 1; m >>= 1) v = fmaxf(v, __shfl_xor(v, m, 32));
  return v;
}

// ---------------- zero density map ----------------
__global__ void zero_dm_kernel(float4* __restrict__ p) {
  unsigned i = blockIdx.x * blockDim.x + threadIdx.x;
  float4 z; z.x = 0.f; z.y = 0.f; z.z = 0.f; z.w = 0.f;
  p[i] = z;
}

// ---------------- bilinear scatter-add ----------------
__device__ __forceinline__ void deposit(float* __restrict__ dmb, float r, float c) {
  float xf = fminf(fmaxf(floorf(r), 0.0f), 511.0f);
  float yf = fminf(fmaxf(floorf(c), 0.0f), 511.0f);
  float xc = fminf(fmaxf(ceilf(r),  0.0f), 511.0f);
  float yc = fminf(fmaxf(ceilf(c),  0.0f), 511.0f);
  float xw = r - xf;
  float yw = c - yf;
  int ixf = (int)xf, iyf = (int)yf, ixc = (int)xc, iyc = (int)yc;
  atomicAdd(dmb + ixf * A_DIM + iyf, (1.0f - xw) * (1.0f - yw));
  atomicAdd(dmb + ixc * A_DIM + iyf, xw * (1.0f - yw));
  atomicAdd(dmb + ixf * A_DIM + iyc, (1.0f - xw) * yw);
  atomicAdd(dmb + ixc * A_DIM + iyc, xw * yw);
}

__global__ void scatter_kernel(const float* __restrict__ r3, const float* __restrict__ c3,
                               const float* __restrict__ r4, const float* __restrict__ c4,
                               float* __restrict__ dm) {
  unsigned idx = blockIdx.x * TPB + threadIdx.x;     // 0 .. NE-1 ; batch = idx >> 18
  float* dmb = dm + ((idx >> 18) << 18);             // b * NPIX
  deposit(dmb, r3[idx], c3[idx]);
  deposit(dmb, r4[idx], c4[idx]);
}

// ---------------- max reduction (2-stage, deterministic) ----------------
__global__ void max_partial_kernel(const float* __restrict__ dm, float* __restrict__ part) {
  unsigned tid = blockIdx.x * TPB + threadIdx.x;
  float m = 0.0f;                                    // values are >= 0
#pragma unroll
  for (int k = 0; k < EPT; ++k) m = fmaxf(m, dm[tid + (unsigned)k * NTHREADS]);
  m = wave_max32(m);
  __shared__ float s[TPB / 32];
  if ((threadIdx.x & 31) == 0) s[threadIdx.x >> 5] = m;
  __syncthreads();
  if (threadIdx.x == 0) {
    float r = s[0];
    for (int w = 1; w < TPB / 32; ++w) r = fmaxf(r, s[w]);
    part[blockIdx.x] = r;
  }
}

__global__ void final_max_512(const float* __restrict__ p, float* __restrict__ out) {
  __shared__ float s[256];
  int t = threadIdx.x;
  s[t] = fmaxf(p[t], p[t + 256]);
  __syncthreads();
  for (int off = 128; off > 0; off >>= 1) {
    if (t < off) s[t] = fmaxf(s[t], s[t + off]);
    __syncthreads();
  }
  if (t == 0) out[0] = s[0];
}

// ---------------- count of (target == 1) ----------------
__global__ void count_partial_kernel(const float* __restrict__ tgt, float* __restrict__ part) {
  unsigned tid = blockIdx.x * TPB + threadIdx.x;
  float cnt = 0.0f;
#pragma unroll
  for (int k = 0; k < EPT; ++k) cnt += (tgt[tid + (unsigned)k * NTHREADS] == 1.0f) ? 1.0f : 0.0f;
  cnt = wave_sum32(cnt);
  __shared__ float s[TPB / 32];
  if ((threadIdx.x & 31) == 0) s[threadIdx.x >> 5] = cnt;
  __syncthreads();
  if (threadIdx.x == 0) {
    float r = 0.0f;
    for (int w = 0; w < TPB / 32; ++w) r += s[w];
    part[blockIdx.x] = r;
  }
}

// ---------------- masked BCE partial sums, accumulated with WMMA ----------------
// D = ones(16x4) x B + C  => every row of D holds the column-sums of B, so each
// v_wmma_f32_16x16x4_f32 folds 64 f32 values into the accumulator in full f32.
// Sum of acc[0] over 32 lanes = 2 * total (lanes 0-15: row M=0, lanes 16-31: row M=8).
__global__ void bce_partial_kernel(const float* __restrict__ dm,
                                   const float* __restrict__ tgt,
                                   const float* __restrict__ maxv,
                                   float* __restrict__ part) {
  const unsigned tid = blockIdx.x * TPB + threadIdx.x;
  const float inv = 1.0f / maxv[0];
  v8f acc = {};
  v2f ones; ones[0] = 1.0f; ones[1] = 1.0f;
#pragma unroll
  for (int it = 0; it < EPT / 2; ++it) {
    unsigned i0 = tid + (unsigned)(2 * it) * NTHREADS;
    unsigned i1 = i0 + NTHREADS;
    float p0 = dm[i0] * inv;
    float p1 = dm[i1] * inv;
    float l0 = fmaxf(logf(p0), -100.0f);   // log(0) -> -inf -> clamp to -100
    float l1 = fmaxf(logf(p1), -100.0f);
    float v0 = (tgt[i0] == 1.0f) ? -l0 : 0.0f;
    float v1 = (tgt[i1] == 1.0f) ? -l1 : 0.0f;
    v2f b; b[0] = v0; b[1] = v1;
    acc = __builtin_amdgcn_wmma_f32_16x16x4_f32(
        /*neg_a=*/false, ones, /*neg_b=*/false, b,
        /*c_mod=*/(short)0, acc, /*reuse_a=*/false, /*reuse_b=*/false);
  }
  float wsum = 0.5f * wave_sum32(acc[0]);
  __shared__ float s[TPB / 32];
  if ((threadIdx.x & 31) == 0) s[threadIdx.x >> 5] = wsum;
  __syncthreads();
  if (threadIdx.x == 0) {
    float r = 0.0f;
    for (int w = 0; w < TPB / 32; ++w) r += s[w];
    part[blockIdx.x] = r;
  }
}

__global__ void final_sum_512(const float* __restrict__ p, float* __restrict__ out) {
  __shared__ float s[256];
  int t = threadIdx.x;
  s[t] = p[t] + p[t + 256];
  __syncthreads();
  for (int off = 128; off > 0; off >>= 1) {
    if (t < off) s[t] += s[t + off];
    __syncthreads();
  }
  if (t == 0) out[0] = s[0];
}

// ---------------- final combine: out = (sum of 7 scale sums) / count ----------------
__global__ void combine_kernel(const float* __restrict__ scalars, float* __restrict__ out) {
  float s = 0.0f;
  for (int i = 0; i < NUM_SCALES; ++i) s += scalars[2 + i];
  out[0] = s / scalars[1];
}

extern "C" void kernel_launch(void* const* d_in, const int* in_sizes, int n_in,
                              void* d_out, int out_size, void* d_ws, size_t ws_size,
                              hipStream_t stream) {
  const float* r3  = (const float*)d_in[0];
  const float* c3  = (const float*)d_in[1];
  const float* r4  = (const float*)d_in[2];
  const float* c4  = (const float*)d_in[3];
  const float* tgt = (const float*)d_in[4];

  // workspace layout (floats): dm[NE] | maxPart[512] | bcePart[512] | cntPart[512] | scalars[9]
  float* ws      = (float*)d_ws;
  float* dm      = ws;
  float* maxPart = ws + NE;
  float* bcePart = maxPart + NBLK;
  float* cntPart = bcePart + NBLK;
  float* scalars = cntPart + NBLK;   // [0]=current scale max, [1]=count, [2..8]=per-scale sums

  // count of target==1 (shared by all scales)
  count_partial_kernel<<<NBLK, TPB, 0, stream>>>(tgt, cntPart);
  final_sum_512<<<1, 256, 0, stream>>>(cntPart, scalars + 1);

  for (int i = 0; i < NUM_SCALES; ++i) {
    const size_t off = (size_t)i * NE;
    zero_dm_kernel<<<NE / 4 / TPB, TPB, 0, stream>>>((float4*)dm);
    scatter_kernel<<<NE / TPB, TPB, 0, stream>>>(r3 + off, c3 + off, r4 + off, c4 + off, dm);
    max_partial_kernel<<<NBLK, TPB, 0, stream>>>(dm, maxPart);
    final_max_512<<<1, 256, 0, stream>>>(maxPart, scalars);
    bce_partial_kernel<<<NBLK, TPB, 0, stream>>>(dm, tgt, scalars, bcePart);
    final_sum_512<<<1, 256, 0, stream>>>(bcePart, scalars + 2 + i);
  }

  combine_kernel<<<1, 1, 0, stream>>>(scalars, (float*)d_out);
}